// ResultEncoder_62036507623645
// MI455X (gfx1250) — compile-verified
//
#include <hip/hip_runtime.h>

typedef __attribute__((ext_vector_type(2))) float v2f;
typedef __attribute__((ext_vector_type(4))) float v4f;
typedef __attribute__((ext_vector_type(8))) float v8f;

#if __has_builtin(__builtin_amdgcn_wmma_f32_16x16x4_f32)
#define USE_WMMA_F32 1
#endif

constexpr int kB = 8;          // batch
constexpr int kT = 16384;      // N*S tokens
constexpr int kC = 768;        // channels
constexpr int kCv = kC / 4;    // channels in float4 units
constexpr int kD = 128;
constexpr int kTS = 16;        // token splits
constexpr int kCGV = 6;        // channel groups of 128 (32 lanes x float4)
constexpr int kTOK_WG = kT / kTS;        // 1024 tokens per workgroup
constexpr int kTOK_TH = kTOK_WG / 8;     // 128 tokens per thread

// Branchless-indexed running top-16 (constant indices only -> stays in VGPRs).
__device__ __forceinline__ void topk_insert(float (&t)[16], float& tmin, float v) {
    if (v > tmin) {
        bool done = false;
#pragma unroll
        for (int j = 0; j < 16; ++j) {
            bool rep = (!done) && (t[j] == tmin);
            t[j] = rep ? v : t[j];
            done = done || rep;
        }
        float mv = t[0];
#pragma unroll
        for (int j = 1; j < 16; ++j) mv = fminf(mv, t[j]);
        tmin = mv;
    }
}

// -------- Kernel 1: streaming top-16 (l4) + sum (l12), one fused pass --------
// grid.x = kB * kCGV * kTS ; block = 256 (8 waves).
// Each lane owns 4 consecutive channels -> 512B contiguous per wave per
// global_load_b128; loads are non-temporal (data is touched exactly once and
// l4 alone (402MB) exceeds the 192MB L2).
__global__ __launch_bounds__(256) void k_reduce(const float* __restrict__ l4,
                                                const float* __restrict__ l12,
                                                float* __restrict__ cand,
                                                float* __restrict__ psum) {
    __shared__ float sT[256 * 16];
    __shared__ float sS[256 * 4];
    const int tid = threadIdx.x;
    const int blk = blockIdx.x;
    const int b   = blk / (kCGV * kTS);
    const int rem = blk % (kCGV * kTS);
    const int cg  = rem / kTS;
    const int ts  = rem % kTS;
    const int cl  = tid & 31;        // lane -> channel quad
    const int tc  = tid >> 5;        // wave index -> token sub-chunk
    const int tok0 = ts * kTOK_WG + tc * kTOK_TH;

    const v4f* __restrict__ l4v  = (const v4f*)l4;
    const v4f* __restrict__ l12v = (const v4f*)l12;

    float t[4][16];
    float tmin[4];
    float s12[4];
#pragma unroll
    for (int cc = 0; cc < 4; ++cc) {
        tmin[cc] = -3.4e38f;
        s12[cc] = 0.f;
#pragma unroll
        for (int j = 0; j < 16; ++j) t[cc][j] = -3.4e38f;
    }

    for (int i = 0; i < kTOK_TH; ++i) {
        size_t idx = ((size_t)b * kT + (size_t)(tok0 + i)) * kCv + cg * 32 + cl;
        v4f a4  = __builtin_nontemporal_load(l4v + idx);
        v4f a12 = __builtin_nontemporal_load(l12v + idx);
#pragma unroll
        for (int cc = 0; cc < 4; ++cc) {
            s12[cc] += a12[cc];
            topk_insert(t[cc], tmin[cc], a4[cc]);
        }
    }

#pragma unroll
    for (int cc = 0; cc < 4; ++cc) sS[tid * 4 + cc] = s12[cc];

    // 4 rounds reusing one 16KB LDS list buffer: round cc merges channel
    // (cg*128 + lane*4 + cc) across the 8 token sub-chunks.
    for (int cc = 0; cc < 4; ++cc) {
#pragma unroll
        for (int j = 0; j < 16; ++j) sT[tid * 16 + j] = t[cc][j];
        __syncthreads();
        if (tid < 32) {
            float m[16];
#pragma unroll
            for (int j = 0; j < 16; ++j) m[j] = -3.4e38f;
            float mmin = -3.4e38f;
            float ss = 0.f;
            for (int k = 0; k < 8; ++k) {
                ss += sS[(k * 32 + tid) * 4 + cc];
#pragma unroll
                for (int j = 0; j < 16; ++j) topk_insert(m, mmin, sT[(k * 32 + tid) * 16 + j]);
            }
            const int ch = cg * 128 + tid * 4 + cc;
            const size_t base = (size_t)(b * kC + ch) * kTS + ts;
#pragma unroll
            for (int j = 0; j < 16; ++j) cand[base * 16 + j] = m[j];
            psum[base] = ss;
        }
        __syncthreads();
    }
}

// -------- Kernel 2: final merge per (b,c); writes zero-padded 16-row x --------
// Threads [0, 6144): real (b,c) work.  Threads [6144, 12288): zero rows 8..15
// of the padded projector inputs so GEMM1's A loads need no predication.
__global__ __launch_bounds__(256) void k_merge(const float* __restrict__ cand,
                                               const float* __restrict__ psum,
                                               float* __restrict__ texpad,
                                               float* __restrict__ strpad) {
    const int g = blockIdx.x * 256 + threadIdx.x;
    if (g < kB * kC) {
        float t[16];
#pragma unroll
        for (int j = 0; j < 16; ++j) t[j] = -3.4e38f;
        float tmin = -3.4e38f;
        for (int i = 0; i < kTS * 16; ++i) topk_insert(t, tmin, cand[(size_t)g * kTS * 16 + i]);
        float s = 0.f;
#pragma unroll
        for (int j = 0; j < 16; ++j) s += t[j];
        texpad[g] = s * (1.0f / 16.0f);
        float ps = 0.f;
        for (int i = 0; i < kTS; ++i) ps += psum[(size_t)g * kTS + i];
        strpad[g] = ps * (1.0f / (float)kT);
    } else if (g < 2 * kB * kC) {
        texpad[g] = 0.f;
        strpad[g] = 0.f;
    }
}

// -------- Kernel 3: both projectors via V_WMMA_F32_16X16X4_F32 --------
// 32 waves: wave = (projector<<4) | n_tile.  M=16 (rows 0..7 real, 8..15 zero).
__global__ __launch_bounds__(1024) void k_proj(
    const float* __restrict__ texpad, const float* __restrict__ strpad,
    const float* __restrict__ tw1, const float* __restrict__ tb1,
    const float* __restrict__ tg1, const float* __restrict__ tbt1,
    const float* __restrict__ tw2, const float* __restrict__ tb2,
    const float* __restrict__ tg2, const float* __restrict__ tbt2,
    const float* __restrict__ sw1, const float* __restrict__ sb1,
    const float* __restrict__ sg1, const float* __restrict__ sbt1,
    const float* __restrict__ sw2, const float* __restrict__ sb2,
    const float* __restrict__ sg2, const float* __restrict__ sbt2,
    float* __restrict__ out) {
    __shared__ float hbuf[2 * 16 * 256];  // post-GEMM1 activations
    __shared__ float obuf[2 * 16 * 128];  // post-GEMM2 activations
    const int tid = threadIdx.x;
    const int wave = tid >> 5;
    const int lane = tid & 31;
    const int r  = lane & 15;   // A-matrix M row / B-matrix N column within tile
    const int kh = lane >> 4;   // K half: lanes 16..31 carry K+2, K+3

    // ---- Stage A: h = x @ w1 + b1   (K=768, 192 WMMA per tile) ----
    {
        const int proj = wave >> 4;
        const int nt = wave & 15;
        const float* x  = proj ? strpad : texpad;   // 16x768, rows 8..15 zero
        const float* w1 = proj ? sw1 : tw1;
        const float* b1 = proj ? sb1 : tb1;
        const int n = nt * 16 + r;
        v8f acc = {};
        for (int k0 = 0; k0 < kC; k0 += 4) {
            const int kk = k0 + 2 * kh;                       // even -> 8B aligned
            v2f a = *(const v2f*)(x + r * kC + kk);           // global_load_b64
            v2f bm;
            bm.x = w1[(size_t)kk * 256 + n];                  // B: K=kk,   N=n
            bm.y = w1[(size_t)(kk + 1) * 256 + n];            // B: K=kk+1, N=n
#ifdef USE_WMMA_F32
            acc = __builtin_amdgcn_wmma_f32_16x16x4_f32(false, a, false, bm,
                                                        (short)0, acc, false, false);
#else
            (void)a; (void)bm;  // builtin absent on this toolchain: no-op stub
#endif
        }
        const float bias = b1[n];
#pragma unroll
        for (int j = 0; j < 8; ++j) {
            const int m = j + kh * 8;  // C/D layout: vgpr j -> M = j (+8 for hi half)
            hbuf[proj * 4096 + m * 256 + n] = acc[j] + bias;
        }
    }
    __syncthreads();

    // ---- Stage B: layernorm(256) + exact-erf GELU (one wave per real row) ----
    if (wave < 16) {
        const int proj = wave >> 3;
        const int row = wave & 7;
        const float* g1  = proj ? sg1 : tg1;
        const float* bt1 = proj ? sbt1 : tbt1;
        float v[8];
        float s = 0.f, q = 0.f;
#pragma unroll
        for (int e = 0; e < 8; ++e) {
            float xv = hbuf[proj * 4096 + row * 256 + lane + 32 * e];
            v[e] = xv; s += xv; q += xv * xv;
        }
#pragma unroll
        for (int off = 16; off > 0; off >>= 1) {
            s += __shfl_xor(s, off, 32);
            q += __shfl_xor(q, off, 32);
        }
        const float mu = s * (1.f / 256.f);
        const float var = q * (1.f / 256.f) - mu * mu;
        const float rstd = rsqrtf(var + 1e-5f);
#pragma unroll
        for (int e = 0; e < 8; ++e) {
            const int n = lane + 32 * e;
            float y = (v[e] - mu) * rstd * g1[n] + bt1[n];
            y = 0.5f * y * (1.f + erff(y * 0.70710678118654752440f));  // exact GELU
            hbuf[proj * 4096 + row * 256 + n] = y;
        }
    } else {  // zero padded rows 8..15 so GEMM2's padded output is benign
        const int proj = (wave - 16) >> 3;
        const int row = 8 + ((wave - 16) & 7);
#pragma unroll
        for (int e = 0; e < 8; ++e) hbuf[proj * 4096 + row * 256 + lane + 32 * e] = 0.f;
    }
    __syncthreads();

    // ---- Stage C: o = h @ w2 + b2   (K=256, 64 WMMA per tile; 8 tiles/projector) ----
    {
        const int proj = wave >> 4;
        const int nt = wave & 15;
        if (nt < 8) {  // wave-uniform branch: EXEC stays all-ones inside
            const float* w2 = proj ? sw2 : tw2;
            const float* b2 = proj ? sb2 : tb2;
            const int n = nt * 16 + r;
            v8f acc = {};
            for (int k0 = 0; k0 < 256; k0 += 4) {
                const int kk = k0 + 2 * kh;                           // even
                v2f a = *(const v2f*)(&hbuf[proj * 4096 + r * 256 + kk]);  // ds_load_b64
                v2f bm;
                bm.x = w2[kk * kD + n];
                bm.y = w2[(kk + 1) * kD + n];
#ifdef USE_WMMA_F32
                acc = __builtin_amdgcn_wmma_f32_16x16x4_f32(false, a, false, bm,
                                                            (short)0, acc, false, false);
#else
                (void)a; (void)bm;
#endif
            }
            const float bias = b2[n];
#pragma unroll
            for (int j = 0; j < 8; ++j) {
                const int m = j + kh * 8;
                obuf[proj * 2048 + m * kD + n] = acc[j] + bias;
            }
        }
    }
    __syncthreads();

    // ---- Stage D: layernorm(128) + l2norm + store ----
    if (wave < 16) {
        const int proj = wave >> 3;
        const int row = wave & 7;
        const float* g2  = proj ? sg2 : tg2;
        const float* bt2 = proj ? sbt2 : tbt2;
        float v[4];
        float s = 0.f, q = 0.f;
#pragma unroll
        for (int e = 0; e < 4; ++e) {
            float xv = obuf[proj * 2048 + row * kD + lane + 32 * e];
            v[e] = xv; s += xv; q += xv * xv;
        }
#pragma unroll
        for (int off = 16; off > 0; off >>= 1) {
            s += __shfl_xor(s, off, 32);
            q += __shfl_xor(q, off, 32);
        }
        const float mu = s * (1.f / 128.f);
        const float var = q * (1.f / 128.f) - mu * mu;
        const float rstd = rsqrtf(var + 1e-5f);
        float y[4];
        float sq = 0.f;
#pragma unroll
        for (int e = 0; e < 4; ++e) {
            const int n = lane + 32 * e;
            float tv = (v[e] - mu) * rstd * g2[n] + bt2[n];
            y[e] = tv; sq += tv * tv;
        }
#pragma unroll
        for (int off = 16; off > 0; off >>= 1) sq += __shfl_xor(sq, off, 32);
        const float inv = 1.f / fmaxf(sqrtf(sq), 1e-12f);
#pragma unroll
        for (int e = 0; e < 4; ++e)
            out[proj * (8 * kD) + row * kD + lane + 32 * e] = y[e] * inv;
    }
}

extern "C" void kernel_launch(void* const* d_in, const int* in_sizes, int n_in,
                              void* d_out, int out_size, void* d_ws, size_t ws_size,
                              hipStream_t stream) {
    const float* l4   = (const float*)d_in[0];
    const float* l12  = (const float*)d_in[1];
    const float* tw1  = (const float*)d_in[2];
    const float* tb1  = (const float*)d_in[3];
    const float* tg1  = (const float*)d_in[4];
    const float* tbt1 = (const float*)d_in[5];
    const float* tw2  = (const float*)d_in[6];
    const float* tb2  = (const float*)d_in[7];
    const float* tg2  = (const float*)d_in[8];
    const float* tbt2 = (const float*)d_in[9];
    const float* sw1  = (const float*)d_in[10];
    const float* sb1  = (const float*)d_in[11];
    const float* sg1  = (const float*)d_in[12];
    const float* sbt1 = (const float*)d_in[13];
    const float* sw2  = (const float*)d_in[14];
    const float* sb2  = (const float*)d_in[15];
    const float* sg2  = (const float*)d_in[16];
    const float* sbt2 = (const float*)d_in[17];

    float* ws = (float*)d_ws;
    float* cand   = ws;                                       // 8*768*16*16 = 1,572,864 f
    float* psum   = cand   + (size_t)kB * kC * kTS * 16;      //   8*768*16 =    98,304 f
    float* texpad = psum   + (size_t)kB * kC * kTS;           //     16*768 =    12,288 f
    float* strpad = texpad + (size_t)2 * kB * kC;             //     16*768 =    12,288 f

    k_reduce<<<kB * kCGV * kTS, 256, 0, stream>>>(l4, l12, cand, psum);
    k_merge<<<(2 * kB * kC + 255) / 256, 256, 0, stream>>>(cand, psum, texpad, strpad);
    k_proj<<<1, 1024, 0, stream>>>(texpad, strpad,
                                   tw1, tb1, tg1, tbt1, tw2, tb2, tg2, tbt2,
                                   sw1, sb1, sg1, sbt1, sw2, sb2, sg2, sbt2,
                                   (float*)d_out);
}